// HeteroClassifier_8924942041219
// MI455X (gfx1250) — compile-verified
//
#include <hip/hip_runtime.h>
#include <hip/hip_bf16.h>

// ---------------- problem constants ----------------
#define NN 200000      // nodes
#define RR 4           // relations
#define EE 1000000     // edges per relation
#define BB 1024        // graphs

typedef float v2f __attribute__((ext_vector_type(2)));
typedef float v8f __attribute__((ext_vector_type(8)));

// ---------------- workspace layout (float offsets) ----------------
#define O_RDOUT 0                          // [R][N] rsqrt(out_deg)
#define O_RDIN  (RR*NN)                    // [R][N] rsqrt(in_deg)
#define O_AGG1  (2*RR*NN)                  // [R][N][2] layer-1 scatter accum
#define O_H     (O_AGG1 + RR*NN*2)         // [N][16] layer-1 output (post relu)
#define O_G     (O_H + NN*16)              // [N][16] h @ W2[r] (reused per relation)
#define O_H2    (O_G + NN*16)              // [N][16] layer-2 accum (all relations)
#define O_GSUM  (O_H2 + NN*16)             // [B][16]
#define O_GCNT  (O_GSUM + BB*16)           // [B]
#define WS_TOTAL (O_GCNT + BB)             // ~12.8M floats (~51MB)

// ---------------- kernels ----------------

__global__ void hc_zero(float* __restrict__ p, int n) {
    for (int i = blockIdx.x * blockDim.x + threadIdx.x; i < n; i += gridDim.x * blockDim.x)
        p[i] = 0.0f;
}

// per-relation degree counts (accumulated as float)
__global__ void hc_degree(const int* __restrict__ src, const int* __restrict__ dst,
                          float* __restrict__ ws) {
    int i = blockIdx.x * blockDim.x + threadIdx.x;
    if (i >= RR * EE) return;
    int r = i / EE;
    atomicAdd(&ws[O_RDOUT + r * NN + src[i]], 1.0f);
    atomicAdd(&ws[O_RDIN  + r * NN + dst[i]], 1.0f);
}

// deg -> rsqrt(max(deg,1)) in place (covers both RDOUT and RDIN ranges)
__global__ void hc_rsqrt(float* __restrict__ ws) {
    int i = blockIdx.x * blockDim.x + threadIdx.x;
    if (i >= 2 * RR * NN) return;
    float v = ws[i];
    v = v < 1.0f ? 1.0f : v;
    ws[i] = rsqrtf(v);
}

// layer-1 edges: scatter 2-vector with both norms + edge weight applied per edge
__global__ void hc_l1_edges(const int* __restrict__ src, const int* __restrict__ dst,
                            const float* __restrict__ ew, const float* __restrict__ feat,
                            float* __restrict__ ws) {
    int i = blockIdx.x * blockDim.x + threadIdx.x;
    if (i >= RR * EE) return;
    int r = i / EE;
    int s = src[i], d = dst[i];
    float w = ew[i] * ws[O_RDOUT + r * NN + s] * ws[O_RDIN + r * NN + d];
    const float2 f = ((const float2*)feat)[s];
    atomicAdd(&ws[O_AGG1 + (r * NN + d) * 2 + 0], f.x * w);
    atomicAdd(&ws[O_AGG1 + (r * NN + d) * 2 + 1], f.y * w);
}

// layer-1 combine: h[n] = relu( sum_r agg1_r[n] @ W1[r] + sum_r b1[r] )
__global__ void hc_l1_combine(const float* __restrict__ W1, const float* __restrict__ b1,
                              float* __restrict__ ws) {
    __shared__ float sW1[RR * 2 * 16];
    __shared__ float sB1[16];
    int tid = threadIdx.x;
    if (tid < RR * 2 * 16) sW1[tid] = W1[tid];
    if (tid < 16) sB1[tid] = b1[tid] + b1[16 + tid] + b1[32 + tid] + b1[48 + tid];
    __syncthreads();

    int n = blockIdx.x * blockDim.x + tid;
    if (n >= NN) return;
    float a[2 * RR];
#pragma unroll
    for (int r = 0; r < RR; ++r) {
        a[2 * r + 0] = ws[O_AGG1 + (r * NN + n) * 2 + 0];
        a[2 * r + 1] = ws[O_AGG1 + (r * NN + n) * 2 + 1];
    }
    float out[16];
#pragma unroll
    for (int j = 0; j < 16; ++j) {
        float acc = sB1[j];
#pragma unroll
        for (int r = 0; r < RR; ++r) {
            acc += a[2 * r + 0] * sW1[(2 * r + 0) * 16 + j];
            acc += a[2 * r + 1] * sW1[(2 * r + 1) * 16 + j];
        }
        out[j] = acc > 0.0f ? acc : 0.0f;
    }
    float4* hv = (float4*)(ws + O_H + (size_t)n * 16);
#pragma unroll
    for (int q = 0; q < 4; ++q)
        hv[q] = make_float4(out[4 * q], out[4 * q + 1], out[4 * q + 2], out[4 * q + 3]);
}

// layer-2 pre-transform via WMMA: g = h @ W2[r], one wave per 16-node tile
// Chains four v_wmma_f32_16x16x4_f32 over K=16.
__global__ void hc_l2_transform(const float* __restrict__ hbuf,
                                const float* __restrict__ W2r,
                                float* __restrict__ g) {
    int gtid = blockIdx.x * blockDim.x + threadIdx.x;
    int tile = gtid >> 5;                 // uniform within a wave
    if (tile >= NN / 16) return;          // whole wave exits together -> EXEC all-1s below
    int lane = threadIdx.x & 31;
    int half = lane >> 4;                 // 0: lanes 0-15, 1: lanes 16-31
    int l    = lane & 15;
    int n0   = tile * 16;

    v8f acc = {};
#pragma unroll
    for (int k0 = 0; k0 < 16; k0 += 4) {
        int ka = k0 + 2 * half;
        v2f a, b;
        // A 16x4 f32: lane half selects K pair {k0,k0+1} / {k0+2,k0+3}, row M = l
        a.x = hbuf[(n0 + l) * 16 + ka + 0];
        a.y = hbuf[(n0 + l) * 16 + ka + 1];
        // B 4x16 f32: row K striped across lanes (mirrored layout), col N = l
        b.x = W2r[(ka + 0) * 16 + l];
        b.y = W2r[(ka + 1) * 16 + l];
        acc = __builtin_amdgcn_wmma_f32_16x16x4_f32(
            /*neg_a=*/false, a, /*neg_b=*/false, b,
            /*c_mod=*/(short)0, acc, /*reuse_a=*/false, /*reuse_b=*/false);
    }
    // D 16x16 f32: VGPR v -> row M=v (lanes 0-15) / M=v+8 (lanes 16-31), col N = l
#pragma unroll
    for (int v = 0; v < 8; ++v)
        g[(n0 + v + 8 * half) * 16 + l] = acc[v];
}

// layer-2 edges: h2[d] += g[s] * rsqrt(out_deg_r[s]) * rsqrt(in_deg_r[d])
__global__ void hc_l2_edges(const int* __restrict__ src_r, const int* __restrict__ dst_r,
                            const float* __restrict__ rdout_r, const float* __restrict__ rdin_r,
                            const float* __restrict__ g, float* __restrict__ h2) {
    int e = blockIdx.x * blockDim.x + threadIdx.x;
    if (e >= EE) return;
    int s = src_r[e], d = dst_r[e];
    float w = rdout_r[s] * rdin_r[d];
    const float4* gs = (const float4*)(g + (size_t)s * 16);
    float* hd = h2 + (size_t)d * 16;
#pragma unroll
    for (int q = 0; q < 4; ++q) {
        float4 v = gs[q];
        atomicAdd(&hd[4 * q + 0], v.x * w);
        atomicAdd(&hd[4 * q + 1], v.y * w);
        atomicAdd(&hd[4 * q + 2], v.z * w);
        atomicAdd(&hd[4 * q + 3], v.w * w);
    }
}

// per-graph sums + counts (bias folded in per node)
__global__ void hc_readout1(const int* __restrict__ graph_ids, const float* __restrict__ b2,
                            float* __restrict__ ws) {
    __shared__ float sB2[16];
    int tid = threadIdx.x;
    if (tid < 16) sB2[tid] = b2[tid] + b2[16 + tid] + b2[32 + tid] + b2[48 + tid];
    __syncthreads();

    int n = blockIdx.x * blockDim.x + tid;
    if (n >= NN) return;
    int gi = graph_ids[n];
    atomicAdd(&ws[O_GCNT + gi], 1.0f);
    const float* h2 = ws + O_H2 + (size_t)n * 16;
    float* gs = ws + O_GSUM + (size_t)gi * 16;
#pragma unroll
    for (int j = 0; j < 16; ++j)
        atomicAdd(&gs[j], h2[j] + sB2[j]);
}

// mean + classifier: out[b] = (gsum[b]/max(cnt,1)) @ Wc + bc
__global__ void hc_readout2(const float* __restrict__ Wc, const float* __restrict__ bc,
                            const float* __restrict__ ws, float* __restrict__ out) {
    int b = blockIdx.x * blockDim.x + threadIdx.x;
    if (b >= BB) return;
    float c = ws[O_GCNT + b];
    c = c < 1.0f ? 1.0f : c;
    float inv = 1.0f / c;
    float o0 = bc[0], o1 = bc[1];
    const float* gs = ws + O_GSUM + (size_t)b * 16;
#pragma unroll
    for (int j = 0; j < 16; ++j) {
        float v = gs[j] * inv;
        o0 += v * Wc[2 * j + 0];
        o1 += v * Wc[2 * j + 1];
    }
    out[2 * b + 0] = o0;
    out[2 * b + 1] = o1;
}

// ---------------- launch ----------------
extern "C" void kernel_launch(void* const* d_in, const int* in_sizes, int n_in,
                              void* d_out, int out_size, void* d_ws, size_t ws_size,
                              hipStream_t stream) {
    const float* feat      = (const float*)d_in[0];
    const int*   src       = (const int*)  d_in[1];
    const int*   dst       = (const int*)  d_in[2];
    const float* ew        = (const float*)d_in[3];
    const int*   graph_ids = (const int*)  d_in[4];
    const float* W1        = (const float*)d_in[5];
    const float* b1        = (const float*)d_in[6];
    const float* W2        = (const float*)d_in[7];
    const float* b2        = (const float*)d_in[8];
    const float* Wc        = (const float*)d_in[9];
    const float* bc        = (const float*)d_in[10];
    float* out = (float*)d_out;
    float* ws  = (float*)d_ws;

    const int T = 256;

    // 0) zero accumulators (entire workspace; deterministic per call)
    hc_zero<<<2048, T, 0, stream>>>(ws, WS_TOTAL);

    // 1) per-relation degrees, then rsqrt(clip(.,1))
    hc_degree<<<(RR * EE + T - 1) / T, T, 0, stream>>>(src, dst, ws);
    hc_rsqrt<<<(2 * RR * NN + T - 1) / T, T, 0, stream>>>(ws);

    // 2) layer 1: edge scatter (2 floats/edge) then combine + relu
    hc_l1_edges<<<(RR * EE + T - 1) / T, T, 0, stream>>>(src, dst, ew, feat, ws);
    hc_l1_combine<<<(NN + T - 1) / T, T, 0, stream>>>(W1, b1, ws);

    // 3) layer 2: per relation, WMMA pre-transform g = h @ W2[r], then edge scatter into h2
    const int tiles = NN / 16;
    const int tblocks = (tiles * 32 + T - 1) / T;
    for (int r = 0; r < RR; ++r) {
        hc_l2_transform<<<tblocks, T, 0, stream>>>(ws + O_H, W2 + (size_t)r * 256, ws + O_G);
        hc_l2_edges<<<(EE + T - 1) / T, T, 0, stream>>>(
            src + (size_t)r * EE, dst + (size_t)r * EE,
            ws + O_RDOUT + (size_t)r * NN, ws + O_RDIN + (size_t)r * NN,
            ws + O_G, ws + O_H2);
    }

    // 4) readout: per-graph mean + classifier
    hc_readout1<<<(NN + T - 1) / T, T, 0, stream>>>(graph_ids, b2, ws);
    hc_readout2<<<(BB + T - 1) / T, T, 0, stream>>>(Wc, bc, ws, out);
}